// Network_64579128263053
// MI455X (gfx1250) — compile-verified
//
#include <hip/hip_runtime.h>

// ---------------------------------------------------------------------------
// GAT x2 + MLP over wire-crossing graph, MI455X (gfx1250, wave32).
// Per-replica processing keeps ~31MB working set resident in 192MB L2.
// Feature transforms use V_WMMA_F32_16X16X4_F32 (16-node tiles per wave).
// ---------------------------------------------------------------------------

typedef __attribute__((ext_vector_type(2))) float v2f;
typedef __attribute__((ext_vector_type(8))) float v8f;

#define N_NODES   150000
#define N_EDGES   1800000
#define NW        480
#define NFEATT    4
#define BT        16
#define NEG_SLOPE 0.2f
#define NTILES    (N_NODES / 16)       // 9375 exact

__device__ __forceinline__ float lrelu(float x) { return x > 0.f ? x : NEG_SLOPE * x; }
__device__ __forceinline__ float eluf (float x) { return x > 0.f ? x : (__expf(x) - 1.f); }

// Float atomic max via sign-split int/uint atomics (native global_atomic_max/min,
// no CAS loop). Canonicalize -0 so the integer ordering trick is exact.
__device__ __forceinline__ void atomicMaxF32(float* addr, float val) {
    if (val == 0.f) val = 0.f;                       // -0 -> +0
    if (val >= 0.f) atomicMax((int*)addr, __float_as_int(val));
    else            atomicMin((unsigned int*)addr, (unsigned int)__float_as_int(val));
}

// ---------------------------------------------------------------------------
// Layer-1 transform: gather wire triplet (12 feats) -> h1 = x@W1 (16) via
// 3x V_WMMA_F32_16X16X4_F32, then per-head attention coefficients.
// One wave handles a 16-node tile; 8 waves (tiles) per block.
// ---------------------------------------------------------------------------
__global__ __launch_bounds__(256)
void k_gat1_transform(const float* __restrict__ fw, const float* __restrict__ sw,
                      const float* __restrict__ tw, const int* __restrict__ indices,
                      const float* __restrict__ W1,
                      const float* __restrict__ a1s_w, const float* __restrict__ a1d_w,
                      float* __restrict__ h1, float* __restrict__ a1s, float* __restrict__ a1d)
{
    __shared__ float tile[8][16 * 17];
    const int lane = threadIdx.x & 31;
    const int wave = threadIdx.x >> 5;
    int tileIdx = blockIdx.x * 8 + wave;
    if (tileIdx >= NTILES) tileIdx = NTILES - 1;     // clamp: EXEC stays all-ones
    const int base  = tileIdx * 16;
    const int m     = lane & 15;                     // row within tile / output col
    const int node  = base + m;
    const int khalf = (lane >> 4) << 1;              // 0 or 2 (K-pair per lane half)

    const int i0 = indices[node * 3 + 0];
    const int i1 = indices[node * 3 + 1];
    const int i2 = indices[node * 3 + 2];

    v8f c = {0.f, 0.f, 0.f, 0.f, 0.f, 0.f, 0.f, 0.f};
#pragma unroll
    for (int kb = 0; kb < 12; kb += 4) {
        const int f0    = kb + khalf;                // even feature index
        const int plane = f0 >> 2;                   // wire plane 0/1/2
        const int off   = f0 & 3;                    // 0 or 2 (8B aligned pair)
        const float* wp = (plane == 0) ? fw : ((plane == 1) ? sw : tw);
        const int    wi = (plane == 0) ? i0 : ((plane == 1) ? i1 : i2);
        const float2 av = *(const float2*)(wp + wi * NFEATT + off);
        v2f a; a.x = av.x; a.y = av.y;               // A: 16x4 f32 tile rows
        v2f b;                                       // B: 4x16 f32 (W1 rows f0,f0+1)
        b.x = W1[f0 * 16 + m];
        b.y = W1[(f0 + 1) * 16 + m];
        c = __builtin_amdgcn_wmma_f32_16x16x4_f32(false, a, false, b, (short)0, c,
                                                  false, false);
    }
    // D layout: VGPR r, lanes 0-15 -> M=r, lanes 16-31 -> M=r+8 (§7.12.2)
#pragma unroll
    for (int r = 0; r < 8; ++r)
        tile[wave][(r + 8 * (lane >> 4)) * 17 + m] = c[r];
    __syncthreads();

    // Epilogue: lane handles node m, head hh = lane>>4 (H1=2, D1=8).
    const int hh = lane >> 4;
    float v[8];
#pragma unroll
    for (int d = 0; d < 8; ++d) v[d] = tile[wave][m * 17 + hh * 8 + d];
    float4* hp = (float4*)(h1 + (size_t)node * 16 + hh * 8);
    hp[0] = make_float4(v[0], v[1], v[2], v[3]);
    hp[1] = make_float4(v[4], v[5], v[6], v[7]);
    float as_ = 0.f, ad_ = 0.f;
#pragma unroll
    for (int d = 0; d < 8; ++d) {
        as_ += v[d] * a1s_w[hh * 8 + d];
        ad_ += v[d] * a1d_w[hh * 8 + d];
    }
    a1s[node * 2 + hh] = as_;
    a1d[node * 2 + hh] = ad_;
}

// ---------------------------------------------------------------------------
// Layer-2 transform: x2 = elu(agg1/den1) (16) -> h2 = x2@W2 (4) via
// 4x V_WMMA_F32_16X16X4_F32 (W2 zero-padded to 16 cols), + attention coeffs.
// ---------------------------------------------------------------------------
__global__ __launch_bounds__(256)
void k_gat2_transform(const float* __restrict__ agg1, const float* __restrict__ den1,
                      const float* __restrict__ W2,
                      const float* __restrict__ a2s_w, const float* __restrict__ a2d_w,
                      float* __restrict__ h2, float* __restrict__ a2s, float* __restrict__ a2d)
{
    __shared__ float tile[8][16 * 17];
    const int lane = threadIdx.x & 31;
    const int wave = threadIdx.x >> 5;
    int tileIdx = blockIdx.x * 8 + wave;
    if (tileIdx >= NTILES) tileIdx = NTILES - 1;
    const int base  = tileIdx * 16;
    const int m     = lane & 15;
    const int node  = base + m;
    const int khalf = (lane >> 4) << 1;

    v8f c = {0.f, 0.f, 0.f, 0.f, 0.f, 0.f, 0.f, 0.f};
#pragma unroll
    for (int kb = 0; kb < 16; kb += 4) {
        const int f0 = kb + khalf;                   // even; f0,f0+1 share a head
        const float2 g = *(const float2*)(agg1 + (size_t)node * 16 + f0);
        const float dn = fmaxf(den1[node * 2 + (f0 >> 3)], 1e-16f);
        v2f a; a.x = eluf(g.x / dn); a.y = eluf(g.y / dn);
        v2f b;                                       // W2 is (16,4); pad cols 4..15
        b.x = (m < 4) ? W2[f0 * 4 + m] : 0.f;
        b.y = (m < 4) ? W2[(f0 + 1) * 4 + m] : 0.f;
        c = __builtin_amdgcn_wmma_f32_16x16x4_f32(false, a, false, b, (short)0, c,
                                                  false, false);
    }
#pragma unroll
    for (int r = 0; r < 8; ++r)
        tile[wave][(r + 8 * (lane >> 4)) * 17 + m] = c[r];
    __syncthreads();

    if (lane < 16) {                                 // only cols 0..3 are valid
        float v0 = tile[wave][m * 17 + 0];
        float v1 = tile[wave][m * 17 + 1];
        float v2 = tile[wave][m * 17 + 2];
        float v3 = tile[wave][m * 17 + 3];
        *(float4*)(h2 + (size_t)node * 4) = make_float4(v0, v1, v2, v3);
        a2s[node] = v0 * a2s_w[0] + v1 * a2s_w[1] + v2 * a2s_w[2] + v3 * a2s_w[3];
        a2d[node] = v0 * a2d_w[0] + v1 * a2d_w[1] + v2 * a2d_w[2] + v3 * a2d_w[3];
    }
}

// ---------------------------------------------------------------------------
// Per-replica scratch init (m=-inf, denom/agg=0).
// ---------------------------------------------------------------------------
__global__ __launch_bounds__(256)
void k_init(float* m1, float* den1, float* agg1, float* m2, float* den2, float* agg2)
{
    const int t0     = blockIdx.x * blockDim.x + threadIdx.x;
    const int stride = gridDim.x * blockDim.x;
    const float ninf = -__builtin_inff();
    for (int i = t0; i < N_NODES * 2;  i += stride) { m1[i] = ninf; den1[i] = 0.f; }
    for (int i = t0; i < N_NODES * 16; i += stride) agg1[i] = 0.f;
    for (int i = t0; i < N_NODES;      i += stride) { m2[i] = ninf; den2[i] = 0.f; }
    for (int i = t0; i < N_NODES * 4;  i += stride) agg2[i] = 0.f;
}

__global__ __launch_bounds__(256)
void k_fixmax(float* m, int n)
{
    const int i = blockIdx.x * blockDim.x + threadIdx.x;
    if (i < n) {
        const float v = m[i];
        if (!__builtin_isfinite(v)) m[i] = 0.f;      // no-edge nodes: -inf -> 0
    }
}

// ---------------------------------------------------------------------------
// Layer-1 edge passes (H1=2). alphas gathered per edge end; max/agg atomics
// land in L2-resident node arrays.
// ---------------------------------------------------------------------------
__global__ __launch_bounds__(256)
void k_edge1_max(const int* __restrict__ src, const int* __restrict__ dst,
                 const float* __restrict__ a1s, const float* __restrict__ a1d,
                 float* __restrict__ m1)
{
    const int e = blockIdx.x * blockDim.x + threadIdx.x;
    if (e >= N_EDGES) return;
    const int s = src[e], d = dst[e];
    const float2 as = *(const float2*)(a1s + (size_t)s * 2);
    const float2 ad = *(const float2*)(a1d + (size_t)d * 2);
    atomicMaxF32(&m1[d * 2 + 0], lrelu(as.x + ad.x));
    atomicMaxF32(&m1[d * 2 + 1], lrelu(as.y + ad.y));
}

__global__ __launch_bounds__(256)
void k_edge1_agg(const int* __restrict__ src, const int* __restrict__ dst,
                 const float* __restrict__ a1s, const float* __restrict__ a1d,
                 const float* __restrict__ m1, const float* __restrict__ h1,
                 float* __restrict__ den1, float* __restrict__ agg1)
{
    const int e = blockIdx.x * blockDim.x + threadIdx.x;
    if (e >= N_EDGES) return;
    const int s = src[e], d = dst[e];
    __builtin_prefetch(h1 + (size_t)s * 16, 0, 0);   // global_prefetch_b8
    const float2 as = *(const float2*)(a1s + (size_t)s * 2);
    const float2 ad = *(const float2*)(a1d + (size_t)d * 2);
    const float2 mm = *(const float2*)(m1 + (size_t)d * 2);
    const float w0 = __expf(lrelu(as.x + ad.x) - mm.x);
    const float w1 = __expf(lrelu(as.y + ad.y) - mm.y);
    atomicAdd(&den1[d * 2 + 0], w0);
    atomicAdd(&den1[d * 2 + 1], w1);
    const float4* hs = (const float4*)(h1 + (size_t)s * 16);
    const float4 q0 = hs[0], q1 = hs[1], q2 = hs[2], q3 = hs[3];
    float* o = agg1 + (size_t)d * 16;
    atomicAdd(o + 0,  w0 * q0.x); atomicAdd(o + 1,  w0 * q0.y);
    atomicAdd(o + 2,  w0 * q0.z); atomicAdd(o + 3,  w0 * q0.w);
    atomicAdd(o + 4,  w0 * q1.x); atomicAdd(o + 5,  w0 * q1.y);
    atomicAdd(o + 6,  w0 * q1.z); atomicAdd(o + 7,  w0 * q1.w);
    atomicAdd(o + 8,  w1 * q2.x); atomicAdd(o + 9,  w1 * q2.y);
    atomicAdd(o + 10, w1 * q2.z); atomicAdd(o + 11, w1 * q2.w);
    atomicAdd(o + 12, w1 * q3.x); atomicAdd(o + 13, w1 * q3.y);
    atomicAdd(o + 14, w1 * q3.z); atomicAdd(o + 15, w1 * q3.w);
}

// ---------------------------------------------------------------------------
// Layer-2 edge passes (H2=1, D2=4).
// ---------------------------------------------------------------------------
__global__ __launch_bounds__(256)
void k_edge2_max(const int* __restrict__ src, const int* __restrict__ dst,
                 const float* __restrict__ a2s, const float* __restrict__ a2d,
                 float* __restrict__ m2)
{
    const int e = blockIdx.x * blockDim.x + threadIdx.x;
    if (e >= N_EDGES) return;
    const int s = src[e], d = dst[e];
    atomicMaxF32(&m2[d], lrelu(a2s[s] + a2d[d]));
}

__global__ __launch_bounds__(256)
void k_edge2_agg(const int* __restrict__ src, const int* __restrict__ dst,
                 const float* __restrict__ a2s, const float* __restrict__ a2d,
                 const float* __restrict__ m2, const float* __restrict__ h2,
                 float* __restrict__ den2, float* __restrict__ agg2)
{
    const int e = blockIdx.x * blockDim.x + threadIdx.x;
    if (e >= N_EDGES) return;
    const int s = src[e], d = dst[e];
    __builtin_prefetch(h2 + (size_t)s * 4, 0, 0);
    const float w = __expf(lrelu(a2s[s] + a2d[d]) - m2[d]);
    atomicAdd(&den2[d], w);
    const float4 q = *(const float4*)(h2 + (size_t)s * 4);
    float* o = agg2 + (size_t)d * 4;
    atomicAdd(o + 0, w * q.x); atomicAdd(o + 1, w * q.y);
    atomicAdd(o + 2, w * q.z); atomicAdd(o + 3, w * q.w);
}

// ---------------------------------------------------------------------------
// Final: y = agg2/den2 ; out = y @ mlp_w + mlp_b
// ---------------------------------------------------------------------------
__global__ __launch_bounds__(256)
void k_final(const float* __restrict__ agg2, const float* __restrict__ den2,
             const float* __restrict__ mlp_w, const float* __restrict__ mlp_b,
             float* __restrict__ out)
{
    const int i = blockIdx.x * blockDim.x + threadIdx.x;
    if (i >= N_NODES) return;
    const float dn = fmaxf(den2[i], 1e-16f);
    const float4 y = *(const float4*)(agg2 + (size_t)i * 4);
    const float r = (y.x * mlp_w[0] + y.y * mlp_w[1] +
                     y.z * mlp_w[2] + y.w * mlp_w[3]) / dn;
    out[i] = r + mlp_b[0];
}

// ---------------------------------------------------------------------------
extern "C" void kernel_launch(void* const* d_in, const int* in_sizes, int n_in,
                              void* d_out, int out_size, void* d_ws, size_t ws_size,
                              hipStream_t stream)
{
    const float* fw      = (const float*)d_in[0];
    const float* sw      = (const float*)d_in[1];
    const float* tw      = (const float*)d_in[2];
    const int*   indices = (const int*)  d_in[3];
    const int*   edge    = (const int*)  d_in[4];
    const float* W1      = (const float*)d_in[5];
    const float* a1sw    = (const float*)d_in[6];
    const float* a1dw    = (const float*)d_in[7];
    const float* W2      = (const float*)d_in[8];
    const float* a2sw    = (const float*)d_in[9];
    const float* a2dw    = (const float*)d_in[10];
    const float* mlpw    = (const float*)d_in[11];
    const float* mlpb    = (const float*)d_in[12];
    float*       out     = (float*)d_out;

    // Workspace layout (per-replica, reused across the 16 ticks): ~31 MB.
    float* ws = (float*)d_ws;
    size_t off = 0;
    float* h1   = ws + off; off += (size_t)N_NODES * 16;
    float* a1s  = ws + off; off += (size_t)N_NODES * 2;
    float* a1d  = ws + off; off += (size_t)N_NODES * 2;
    float* m1   = ws + off; off += (size_t)N_NODES * 2;
    float* den1 = ws + off; off += (size_t)N_NODES * 2;
    float* agg1 = ws + off; off += (size_t)N_NODES * 16;
    float* h2   = ws + off; off += (size_t)N_NODES * 4;
    float* a2s  = ws + off; off += (size_t)N_NODES;
    float* a2d  = ws + off; off += (size_t)N_NODES;
    float* m2   = ws + off; off += (size_t)N_NODES;
    float* den2 = ws + off; off += (size_t)N_NODES;
    float* agg2 = ws + off; off += (size_t)N_NODES * 4;

    const int* src = edge;
    const int* dst = edge + N_EDGES;

    const int tileBlocks = (NTILES + 7) / 8;             // 8 waves (tiles) / block
    const int edgeBlocks = (N_EDGES + 255) / 256;
    const int nodeBlocks = (N_NODES + 255) / 256;

    for (int t = 0; t < BT; ++t) {
        const float* fwt = fw + (size_t)t * NW * NFEATT;
        const float* swt = sw + (size_t)t * NW * NFEATT;
        const float* twt = tw + (size_t)t * NW * NFEATT;

        k_init<<<1024, 256, 0, stream>>>(m1, den1, agg1, m2, den2, agg2);
        k_gat1_transform<<<tileBlocks, 256, 0, stream>>>(fwt, swt, twt, indices, W1,
                                                         a1sw, a1dw, h1, a1s, a1d);
        k_edge1_max<<<edgeBlocks, 256, 0, stream>>>(src, dst, a1s, a1d, m1);
        k_fixmax<<<(N_NODES * 2 + 255) / 256, 256, 0, stream>>>(m1, N_NODES * 2);
        k_edge1_agg<<<edgeBlocks, 256, 0, stream>>>(src, dst, a1s, a1d, m1, h1,
                                                    den1, agg1);
        k_gat2_transform<<<tileBlocks, 256, 0, stream>>>(agg1, den1, W2,
                                                         a2sw, a2dw, h2, a2s, a2d);
        k_edge2_max<<<edgeBlocks, 256, 0, stream>>>(src, dst, a2s, a2d, m2);
        k_fixmax<<<nodeBlocks, 256, 0, stream>>>(m2, N_NODES);
        k_edge2_agg<<<edgeBlocks, 256, 0, stream>>>(src, dst, a2s, a2d, m2, h2,
                                                    den2, agg2);
        k_final<<<nodeBlocks, 256, 0, stream>>>(agg2, den2, mlpw, mlpb,
                                                out + (size_t)t * N_NODES);
    }
}